// ForceField_35502199669449
// MI455X (gfx1250) — compile-verified
//
#include <hip/hip_runtime.h>
#include <math.h>

// ---------------- problem sizes (match reference) ----------------
#define NATOM 8192
#define NBOND 8192
#define NANG  12288
#define NUB   8192
#define NDIH  16384
#define NIMP  4096

#define DEG2RAD 0.017453292519943295f

// ---------------- workspace layout (floats) ----------------
// nonbonded grid: 64 x 512 blocks -> 32768 partials
#define NB_GX 64
#define NB_GY 512
#define OFF_NB   0
#define OFF_BOND (OFF_NB + NB_GX * NB_GY)        // 32768, 32 blocks
#define OFF_ANG  (OFF_BOND + 32)                 // 48 blocks
#define OFF_UB   (OFF_ANG + 48)                  // 32 blocks
#define OFF_DIH  (OFF_UB + 32)                   // 64 blocks
#define OFF_IMP  (OFF_DIH + 64)                  // 16 blocks
#define OFF_BOX  (OFF_IMP + 16)                  // 32 blocks
#define NPART    (OFF_BOX + 32)                  // 32992 partial sums total
#define OFF_XSUM NPART                           // 3 floats (sum of coords)
#define OFF_SQ   (OFF_XSUM + 4)                  // 8192 per-atom |x|^2

typedef __attribute__((ext_vector_type(2))) float v2f;
typedef __attribute__((ext_vector_type(8))) float v8f;

// ---------------- block reduction helper ----------------
template <int BS>
__device__ __forceinline__ void block_reduce_store(float acc, float* dst) {
    __shared__ float smem[BS];
    const int t = threadIdx.x;
    smem[t] = acc;
    __syncthreads();
    #pragma unroll
    for (int s = BS >> 1; s > 0; s >>= 1) {
        if (t < s) smem[t] += smem[t + s];
        __syncthreads();
    }
    if (t == 0) *dst = smem[0];
    __syncthreads();
}

// ---------------- per-atom |x|^2 ----------------
__global__ void __launch_bounds__(256)
sq_kernel(const float* __restrict__ x, float* __restrict__ sq) {
    const int i = blockIdx.x * 256 + threadIdx.x;
    if (i < NATOM) {
        const float a = x[3 * i + 0], b = x[3 * i + 1], c = x[3 * i + 2];
        sq[i] = a * a + b * b + c * c;
    }
}

// ---------------- sum of coordinates (for mean) ----------------
__global__ void __launch_bounds__(256)
xsum_kernel(const float* __restrict__ x, float* __restrict__ xsum) {
    float sx = 0.f, sy = 0.f, sz = 0.f;
    for (int i = threadIdx.x; i < NATOM; i += 256) {
        sx += x[3 * i + 0];
        sy += x[3 * i + 1];
        sz += x[3 * i + 2];
    }
    __shared__ float sm[256 * 3];
    const int t = threadIdx.x;
    sm[t] = sx; sm[t + 256] = sy; sm[t + 512] = sz;
    __syncthreads();
    for (int s = 128; s > 0; s >>= 1) {
        if (t < s) {
            sm[t]       += sm[t + s];
            sm[t + 256] += sm[t + 256 + s];
            sm[t + 512] += sm[t + 512 + s];
        }
        __syncthreads();
    }
    if (t == 0) { xsum[0] = sm[0]; xsum[1] = sm[256]; xsum[2] = sm[512]; }
}

// ---------------- nonbonded: WMMA Gram-trick tiles ----------------
// Each wave owns one 16x16 tile of the pair matrix. One
// V_WMMA_F32_16X16X4_F32 produces all 256 dot products x_i . x_j
// (K = x,y,z,0). Then d^2 = |xi|^2 + |xj|^2 - 2*dot, the strict-upper-
// triangle condition is applied as a branchless multiplicative mask,
// and the three parameter matrices are streamed once with NT hints.
__global__ void __launch_bounds__(256)
nb_kernel(const float* __restrict__ x, const float* __restrict__ sq,
          const float* __restrict__ eps, const float* __restrict__ rmin,
          const float* __restrict__ kqq, float* __restrict__ partials) {
    const int lane = threadIdx.x & 31;
    const int w    = threadIdx.x >> 5;
    const int ti   = blockIdx.y;               // tile row (0..511)
    const int tj   = blockIdx.x * 8 + w;       // tile col (0..511)
    const int half = lane >> 4;
    const int p    = lane & 15;

    float acc = 0.f;
    if (tj >= ti) {  // wave-uniform: EXEC stays all-ones around WMMA
        const int i0 = ti * 16, j0 = tj * 16;

        // A (16x4) and B (4x16) fragments share per-lane packing on CDNA5:
        // lanes 0-15 carry K=(0,1)=(x,y); lanes 16-31 carry K=(2,3)=(z,0).
        // Branchless: load all 3 coords, select with v_cndmask.
        v2f a, b;
        {
            const float* xi = x + 3 * (i0 + p);
            const float xr0 = xi[0], xr1 = xi[1], xr2 = xi[2];
            a.x = half ? xr2 : xr0;
            a.y = half ? 0.f : xr1;
            const float* xj = x + 3 * (j0 + p);
            const float xc0 = xj[0], xc1 = xj[1], xc2 = xj[2];
            b.x = half ? xc2 : xc0;
            b.y = half ? 0.f : xc1;
        }
        v8f c = {};
        // D[m][n] = x_{i0+m} . x_{j0+n}
        c = __builtin_amdgcn_wmma_f32_16x16x4_f32(
                /*neg_a=*/false, a, /*neg_b=*/false, b,
                /*c_mod=*/(short)0, c, /*reuse_a=*/false, /*reuse_b=*/false);

        const int   jcol = j0 + p;
        const int   irow = i0 + half * 8;      // first of this lane's 8 rows
        const float sqj  = sq[jcol];

        // One base address per matrix; row stride (32KB) folds into the
        // signed 24-bit instruction offset for r = 0..7.
        const size_t base = (size_t)irow * NATOM + (size_t)jcol;
        const float* pe = eps  + base;
        const float* pr = rmin + base;
        const float* pq = kqq  + base;
        const float* ps = sq   + irow;

        float ev[8], rv[8], qv[8], sv[8];
        #pragma unroll
        for (int r = 0; r < 8; ++r) {          // issue all loads up front
            ev[r] = __builtin_nontemporal_load(pe + (size_t)r * NATOM);
            rv[r] = __builtin_nontemporal_load(pr + (size_t)r * NATOM);
            qv[r] = __builtin_nontemporal_load(pq + (size_t)r * NATOM);
            sv[r] = ps[r];
        }
        #pragma unroll
        for (int r = 0; r < 8; ++r) {
            // C/D layout: VGPR r, lane -> M = r + 8*half, N = p
            const float d2   = sv[r] + sqj - 2.0f * c[r];
            const float inv  = rsqrtf(fmaxf(d2, 1e-24f));   // 1/dist
            const float rod  = rv[r] * inv;
            const float rod3 = rod * rod * rod;
            const float rod6 = rod3 * rod3;
            const float pair = ev[r] * (rod6 * rod6 - 2.0f * rod6) + qv[r] * inv;
            const float m    = (jcol > irow + r) ? 1.0f : 0.0f;  // strict upper tri
            acc = fmaf(m, pair, acc);
        }
    }
    block_reduce_store<256>(acc, partials + blockIdx.y * NB_GX + blockIdx.x);
}

// ---------------- bonds / urey-bradley: k*(|d| - r0)^2 ----------------
__global__ void __launch_bounds__(256)
bond_kernel(const float* __restrict__ x, const int* __restrict__ idx,
            const float* __restrict__ k, const float* __restrict__ r0,
            int n, float* __restrict__ partials) {
    const int i = blockIdx.x * 256 + threadIdx.x;
    float v = 0.f;
    if (i < n) {
        const int a0 = idx[2 * i + 0], a1 = idx[2 * i + 1];
        const float dx = x[3 * a0 + 0] - x[3 * a1 + 0];
        const float dy = x[3 * a0 + 1] - x[3 * a1 + 1];
        const float dz = x[3 * a0 + 2] - x[3 * a1 + 2];
        const float dis = sqrtf(dx * dx + dy * dy + dz * dz);
        const float d   = dis - r0[i];
        v = k[i] * d * d;
    }
    block_reduce_store<256>(v, partials + blockIdx.x);
}

// ---------------- angles: kt*(theta - t0)^2 ----------------
__global__ void __launch_bounds__(256)
angle_kernel(const float* __restrict__ x, const int* __restrict__ ang,
             const float* __restrict__ kt, const float* __restrict__ t0,
             int n, float* __restrict__ partials) {
    const int i = blockIdx.x * 256 + threadIdx.x;
    float v = 0.f;
    if (i < n) {
        const int a = ang[3 * i + 0], b = ang[3 * i + 1], c = ang[3 * i + 2];
        const float bax = x[3 * a + 0] - x[3 * b + 0];
        const float bay = x[3 * a + 1] - x[3 * b + 1];
        const float baz = x[3 * a + 2] - x[3 * b + 2];
        const float bcx = x[3 * c + 0] - x[3 * b + 0];
        const float bcy = x[3 * c + 1] - x[3 * b + 1];
        const float bcz = x[3 * c + 2] - x[3 * b + 2];
        const float dot = bax * bcx + bay * bcy + baz * bcz;
        const float den = sqrtf(bax * bax + bay * bay + baz * baz) *
                          sqrtf(bcx * bcx + bcy * bcy + bcz * bcz);
        float cosv = dot / den;
        cosv = fminf(fmaxf(cosv, -0.9999f), 0.9999f);
        const float d = acosf(cosv) - t0[i] * DEG2RAD;
        v = kt[i] * d * d;
    }
    block_reduce_store<256>(v, partials + blockIdx.x);
}

// ---------------- signed torsion angle ----------------
__device__ __forceinline__ float torsion(const float* __restrict__ x,
                                         int q0, int q1, int q2, int q3) {
    const float abx = x[3 * q1 + 0] - x[3 * q0 + 0];
    const float aby = x[3 * q1 + 1] - x[3 * q0 + 1];
    const float abz = x[3 * q1 + 2] - x[3 * q0 + 2];
    const float bcx = x[3 * q2 + 0] - x[3 * q1 + 0];
    const float bcy = x[3 * q2 + 1] - x[3 * q1 + 1];
    const float bcz = x[3 * q2 + 2] - x[3 * q1 + 2];
    const float cdx = x[3 * q3 + 0] - x[3 * q2 + 0];
    const float cdy = x[3 * q3 + 1] - x[3 * q2 + 1];
    const float cdz = x[3 * q3 + 2] - x[3 * q2 + 2];
    const float n1x = aby * bcz - abz * bcy;
    const float n1y = abz * bcx - abx * bcz;
    const float n1z = abx * bcy - aby * bcx;
    const float n2x = bcy * cdz - bcz * cdy;
    const float n2y = bcz * cdx - bcx * cdz;
    const float n2z = bcx * cdy - bcy * cdx;
    const float dot = n1x * n2x + n1y * n2y + n1z * n2z;
    const float den = sqrtf(n1x * n1x + n1y * n1y + n1z * n1z) *
                      sqrtf(n2x * n2x + n2y * n2y + n2z * n2z);
    float cosv = dot / den;
    cosv = fminf(fmaxf(cosv, -0.9999f), 0.9999f);
    const float ang = acosf(cosv);
    const float cx = n1y * n2z - n1z * n2y;
    const float cy = n1z * n2x - n1x * n2z;
    const float cz = n1x * n2y - n1y * n2x;
    const float s  = cx * bcx + cy * bcy + cz * bcz;
    const float sg = (s > 0.f) ? 1.f : ((s < 0.f) ? -1.f : 0.f);
    return ang * sg;
}

// ---------------- dihedrals: kd*(1 + cos(nd*delta - d0)) ----------------
__global__ void __launch_bounds__(256)
dihedral_kernel(const float* __restrict__ x, const int* __restrict__ q,
                const float* __restrict__ kd, const float* __restrict__ nd,
                const float* __restrict__ d0, int n, float* __restrict__ partials) {
    const int i = blockIdx.x * 256 + threadIdx.x;
    float v = 0.f;
    if (i < n) {
        const float delta = torsion(x, q[4 * i + 0], q[4 * i + 1], q[4 * i + 2], q[4 * i + 3]);
        v = kd[i] * (1.0f + cosf(nd[i] * delta - d0[i] * DEG2RAD));
    }
    block_reduce_store<256>(v, partials + blockIdx.x);
}

// ---------------- impropers: kp*(psi - p0)^2 ----------------
__global__ void __launch_bounds__(256)
improper_kernel(const float* __restrict__ x, const int* __restrict__ q,
                const float* __restrict__ kp, const float* __restrict__ p0,
                int n, float* __restrict__ partials) {
    const int i = blockIdx.x * 256 + threadIdx.x;
    float v = 0.f;
    if (i < n) {
        const float psi = torsion(x, q[4 * i + 0], q[4 * i + 1], q[4 * i + 2], q[4 * i + 3]);
        const float d = psi - p0[i] * DEG2RAD;
        v = kp[i] * d * d;
    }
    block_reduce_store<256>(v, partials + blockIdx.x);
}

// ---------------- box restraint ----------------
__global__ void __launch_bounds__(256)
box_kernel(const float* __restrict__ x, const float* __restrict__ xsum,
           float* __restrict__ partials) {
    const int i = blockIdx.x * 256 + threadIdx.x;
    float v = 0.f;
    if (i < NATOM) {
        const float inv_n = 1.0f / (float)NATOM;
        const float dx = x[3 * i + 0] - xsum[0] * inv_n;
        const float dy = x[3 * i + 1] - xsum[1] * inv_n;
        const float dz = x[3 * i + 2] - xsum[2] * inv_n;
        const float c2 = dx * dx + dy * dy + dz * dz;
        if (sqrtf(c2) > 20.0f) v = 0.1f * c2;
    }
    block_reduce_store<256>(v, partials + blockIdx.x);
}

// ---------------- final deterministic reduction ----------------
__global__ void __launch_bounds__(1024)
reduce_kernel(const float* __restrict__ partials, int n, float* __restrict__ out) {
    float s = 0.f;
    for (int i = threadIdx.x; i < n; i += 1024) s += partials[i];
    __shared__ float sm[1024];
    sm[threadIdx.x] = s;
    __syncthreads();
    for (int st = 512; st > 0; st >>= 1) {
        if (threadIdx.x < st) sm[threadIdx.x] += sm[threadIdx.x + st];
        __syncthreads();
    }
    if (threadIdx.x == 0) out[0] = sm[0];
}

extern "C" void kernel_launch(void* const* d_in, const int* in_sizes, int n_in,
                              void* d_out, int out_size, void* d_ws, size_t ws_size,
                              hipStream_t stream) {
    const float* x    = (const float*)d_in[0];
    const int*   bond = (const int*)  d_in[1];
    const float* kb   = (const float*)d_in[2];
    const float* b0   = (const float*)d_in[3];
    const int*   ang  = (const int*)  d_in[4];
    const float* kt   = (const float*)d_in[5];
    const float* t0   = (const float*)d_in[6];
    const int*   ub   = (const int*)  d_in[7];
    const float* ku   = (const float*)d_in[8];
    const float* s0   = (const float*)d_in[9];
    const int*   dih  = (const int*)  d_in[10];
    const float* kd   = (const float*)d_in[11];
    const float* nd   = (const float*)d_in[12];
    const float* d0   = (const float*)d_in[13];
    const int*   imp  = (const int*)  d_in[14];
    const float* kp   = (const float*)d_in[15];
    const float* p0   = (const float*)d_in[16];
    const float* eps  = (const float*)d_in[17];
    const float* rmin = (const float*)d_in[18];
    const float* kqq  = (const float*)d_in[19];
    float* ws  = (float*)d_ws;
    float* out = (float*)d_out;

    sq_kernel   <<<NATOM / 256, 256, 0, stream>>>(x, ws + OFF_SQ);
    xsum_kernel <<<1,           256, 0, stream>>>(x, ws + OFF_XSUM);
    nb_kernel   <<<dim3(NB_GX, NB_GY), 256, 0, stream>>>(x, ws + OFF_SQ, eps, rmin, kqq, ws + OFF_NB);
    bond_kernel <<<NBOND / 256, 256, 0, stream>>>(x, bond, kb, b0, NBOND, ws + OFF_BOND);
    angle_kernel<<<NANG  / 256, 256, 0, stream>>>(x, ang, kt, t0, NANG, ws + OFF_ANG);
    bond_kernel <<<NUB   / 256, 256, 0, stream>>>(x, ub, ku, s0, NUB, ws + OFF_UB);
    dihedral_kernel<<<NDIH / 256, 256, 0, stream>>>(x, dih, kd, nd, d0, NDIH, ws + OFF_DIH);
    improper_kernel<<<NIMP / 256, 256, 0, stream>>>(x, imp, kp, p0, NIMP, ws + OFF_IMP);
    box_kernel  <<<NATOM / 256, 256, 0, stream>>>(x, ws + OFF_XSUM, ws + OFF_BOX);
    reduce_kernel<<<1, 1024, 0, stream>>>(ws, NPART, out);
}